// Satt_66778151518717
// MI455X (gfx1250) — compile-verified
//
#include <hip/hip_runtime.h>
#include <math.h>

typedef __attribute__((ext_vector_type(2))) float v2f;
typedef __attribute__((ext_vector_type(8))) float v8f;

#define N_NODES 8192
#define BATCH   8
#define DIN     32
#define DH      64

// ---------------------------------------------------------------------------
// Kernel 1: scale[i] = adj[i,i] / (sum_j adj[i,j] + sum_k adjk1[k,i])
// One block per row; float4-coalesced streaming read of the 256MB adjacency.
// ---------------------------------------------------------------------------
__global__ __launch_bounds__(256) void scale_kernel(const float* __restrict__ adj,
                                                    const float* __restrict__ adjk1,
                                                    float* __restrict__ scale) {
  __shared__ float red[256];
  const int i = blockIdx.x;
  const float4* row = (const float4*)(adj + (size_t)i * N_NODES);
  float s = 0.f;
#pragma unroll 4
  for (int j = threadIdx.x; j < N_NODES / 4; j += 256) {
    float4 v = row[j];
    s += (v.x + v.y) + (v.z + v.w);
  }
  // fold adjk1 column-sum (16 values) into the first 16 lanes' partials
  if (threadIdx.x < 16) s += adjk1[threadIdx.x * N_NODES + i];
  red[threadIdx.x] = s;
  __syncthreads();
#pragma unroll
  for (int off = 128; off > 0; off >>= 1) {
    if ((int)threadIdx.x < off) red[threadIdx.x] += red[threadIdx.x + off];
    __syncthreads();
  }
  if (threadIdx.x == 0) {
    scale[i] = adj[(size_t)i * N_NODES + i] / red[0];
  }
}

// ---------------------------------------------------------------------------
// Kernel 2: fused 3-layer gated MLP using V_WMMA_F32_16X16X4_F32.
// 256 threads = 8 waves; each wave owns a 16-node tile (full f32 precision).
// ---------------------------------------------------------------------------
__global__ __launch_bounds__(256) void mlp_kernel(
    const float* __restrict__ x,      // [B, N, 32]
    const float* __restrict__ scale,  // [N]
    const float* __restrict__ W1,  const float* __restrict__ b1,
    const float* __restrict__ W11, const float* __restrict__ b11,
    const float* __restrict__ W2,  const float* __restrict__ b2,
    const float* __restrict__ W22, const float* __restrict__ b22,
    const float* __restrict__ W3,  const float* __restrict__ b3,
    const float* __restrict__ W33, const float* __restrict__ b33,
    float* __restrict__ out)          // [B, N, 64]
{
  __shared__ float hbuf[8][16 * DH];  // per-wave private h-tile scratch (32KB)

  const int wave  = threadIdx.x >> 5;
  const int lane  = threadIdx.x & 31;
  const int mrow  = lane & 15;   // A-frag row / C-frag column
  const int khalf = lane >> 4;   // lane-half selects K pair / M+8
  const int ncol  = mrow;

  const int tile  = blockIdx.x * 8 + wave;
  const int r0    = tile * 16;
  const int b     = r0 / N_NODES;
  const int node0 = r0 % N_NODES;

  float* hl = hbuf[wave];

  // scale for C/D rows M = v + 8*khalf
  float sv[8];
#pragma unroll
  for (int v = 0; v < 8; ++v) sv[v] = scale[node0 + v + 8 * khalf];

  v8f accT[4], accS[4];

  // ======================= layer 1 (K = 32, A from global) =================
#pragma unroll
  for (int nb = 0; nb < 4; ++nb) {
    accT[nb] = (v8f){0.f,0.f,0.f,0.f,0.f,0.f,0.f,0.f};
    accS[nb] = (v8f){0.f,0.f,0.f,0.f,0.f,0.f,0.f,0.f};
  }
  {
    const float* xrow = x + ((size_t)b * N_NODES + node0 + mrow) * DIN;
#pragma unroll
    for (int kb = 0; kb < DIN / 4; ++kb) {
      const int k0 = kb * 4 + 2 * khalf;
      v2f a; a.x = xrow[k0]; a.y = xrow[k0 + 1];
#pragma unroll
      for (int nb = 0; nb < 4; ++nb) {
        const int col = nb * 16 + ncol;
        v2f bt, bs;
        bt.x = W1 [(k0 + 0) * DH + col]; bt.y = W1 [(k0 + 1) * DH + col];
        bs.x = W11[(k0 + 0) * DH + col]; bs.y = W11[(k0 + 1) * DH + col];
        accT[nb] = __builtin_amdgcn_wmma_f32_16x16x4_f32(false, a, false, bt,
                     (short)0, accT[nb], false, false);
        accS[nb] = __builtin_amdgcn_wmma_f32_16x16x4_f32(false, a, false, bs,
                     (short)0, accS[nb], false, false);
      }
    }
  }
  // activation: h = tanh(s*pT + bT) * sigmoid(s*pS + bS)  -> LDS (row-major)
#pragma unroll
  for (int nb = 0; nb < 4; ++nb) {
    const int col = nb * 16 + ncol;
    const float bt = b1[col], bs = b11[col];
#pragma unroll
    for (int v = 0; v < 8; ++v) {
      float pt = sv[v] * accT[nb][v] + bt;
      float ps = sv[v] * accS[nb][v] + bs;
      hl[(v + 8 * khalf) * DH + col] = tanhf(pt) * (1.f / (1.f + expf(-ps)));
    }
  }

  // ================== layers 2 and 3 (K = 64, A from LDS) ==================
#pragma unroll 1
  for (int layer = 0; layer < 2; ++layer) {
    const float* Wt = (layer == 0) ? W2  : W3;
    const float* Ws = (layer == 0) ? W22 : W33;
    const float* ct = (layer == 0) ? b2  : b3;
    const float* cs = (layer == 0) ? b22 : b33;

#pragma unroll
    for (int nb = 0; nb < 4; ++nb) {
      accT[nb] = (v8f){0.f,0.f,0.f,0.f,0.f,0.f,0.f,0.f};
      accS[nb] = (v8f){0.f,0.f,0.f,0.f,0.f,0.f,0.f,0.f};
    }
#pragma unroll
    for (int kb = 0; kb < DH / 4; ++kb) {
      const int k0 = kb * 4 + 2 * khalf;
      v2f a; a.x = hl[mrow * DH + k0]; a.y = hl[mrow * DH + k0 + 1];
#pragma unroll
      for (int nb = 0; nb < 4; ++nb) {
        const int col = nb * 16 + ncol;
        v2f bt, bs;
        bt.x = Wt[(k0 + 0) * DH + col]; bt.y = Wt[(k0 + 1) * DH + col];
        bs.x = Ws[(k0 + 0) * DH + col]; bs.y = Ws[(k0 + 1) * DH + col];
        accT[nb] = __builtin_amdgcn_wmma_f32_16x16x4_f32(false, a, false, bt,
                     (short)0, accT[nb], false, false);
        accS[nb] = __builtin_amdgcn_wmma_f32_16x16x4_f32(false, a, false, bs,
                     (short)0, accS[nb], false, false);
      }
    }

    if (layer == 0) {
      // write h2 back to LDS for layer 3
#pragma unroll
      for (int nb = 0; nb < 4; ++nb) {
        const int col = nb * 16 + ncol;
        const float bt = ct[col], bs = cs[col];
#pragma unroll
        for (int v = 0; v < 8; ++v) {
          float pt = sv[v] * accT[nb][v] + bt;
          float ps = sv[v] * accS[nb][v] + bs;
          hl[(v + 8 * khalf) * DH + col] = tanhf(pt) * (1.f / (1.f + expf(-ps)));
        }
      }
    } else {
      // final activation -> global output [B, N, 64]
#pragma unroll
      for (int nb = 0; nb < 4; ++nb) {
        const int col = nb * 16 + ncol;
        const float bt = ct[col], bs = cs[col];
#pragma unroll
        for (int v = 0; v < 8; ++v) {
          float pt = sv[v] * accT[nb][v] + bt;
          float ps = sv[v] * accS[nb][v] + bs;
          out[((size_t)b * N_NODES + node0 + v + 8 * khalf) * DH + col] =
              tanhf(pt) * (1.f / (1.f + expf(-ps)));
        }
      }
    }
  }
}

// ---------------------------------------------------------------------------
extern "C" void kernel_launch(void* const* d_in, const int* in_sizes, int n_in,
                              void* d_out, int out_size, void* d_ws, size_t ws_size,
                              hipStream_t stream) {
  (void)in_sizes; (void)n_in; (void)out_size; (void)ws_size;
  const float* x     = (const float*)d_in[0];
  const float* adj   = (const float*)d_in[1];
  const float* adjk1 = (const float*)d_in[2];
  const float* W1  = (const float*)d_in[3];
  const float* b1  = (const float*)d_in[4];
  const float* W11 = (const float*)d_in[5];
  const float* b11 = (const float*)d_in[6];
  const float* W2  = (const float*)d_in[7];
  const float* b2  = (const float*)d_in[8];
  const float* W22 = (const float*)d_in[9];
  const float* b22 = (const float*)d_in[10];
  const float* W3  = (const float*)d_in[11];
  const float* b3  = (const float*)d_in[12];
  const float* W33 = (const float*)d_in[13];
  const float* b33 = (const float*)d_in[14];
  float* out   = (float*)d_out;
  float* scale = (float*)d_ws;   // N floats of scratch

  scale_kernel<<<N_NODES, 256, 0, stream>>>(adj, adjk1, scale);

  const int tiles_per_block = 8;                       // 8 waves/block
  const int nblocks = (BATCH * N_NODES) / (16 * tiles_per_block);  // 512
  mlp_kernel<<<nblocks, 256, 0, stream>>>(x, scale,
      W1, b1, W11, b11, W2, b2, W22, b22, W3, b3, W33, b33, out);
}